// DeformableBlock_33200097198274
// MI455X (gfx1250) — compile-verified
//
#include <hip/hip_runtime.h>
#include <math.h>

// ---------------------------------------------------------------------------
// Fused deformable conv block for MI455X (gfx1250, wave32, WMMA).
//   B=8, C_in=64, C_out=64, H=W=96, K=3, pad=1  (fp32 throughout)
// One block = 16 consecutive pixels of one row. Phases:
//   I   build regular im2col panel in LDS, K-pair-interleaved [k/2][16][2]
//   II  offset conv on matrix pipe: waves 0-1, 144x v_wmma_f32_16x16x4_f32
//   III bilinear corner indices/weights for 9 taps x 16 pixels
//   IV  data-dependent gather of sampled panel (overwrites im2col)
//   V   main GEMM: 4 waves x 144 WMMAs, bias, b128 stores
// K-pair interleaving makes every WMMA A-fragment one ds_load_b64 and every
// B-fragment one global_load_b64 (fragments consume K-pairs (even,odd)).
// ---------------------------------------------------------------------------

typedef __attribute__((ext_vector_type(2))) float v2f;
typedef __attribute__((ext_vector_type(8))) float v8f;

#define BB    8
#define CIN   64
#define COUT  64
#define HH    96
#define WW    96
#define K2    9
#define KDIM  (CIN * K2)      // 576 reduction length
#define HW    (HH * WW)       // 9216
#define NOFF  (2 * K2)        // 18 offset channels
#define NOFFP 32              // padded offset channels (2 WMMA N-tiles)
#define MTILE 16              // pixels per block
#define NPAIR (K2 * MTILE)    // 144 (tap, pixel) coordinate sets

// pair-interleaved A-panel address: [k/2][m][2]
__device__ __forceinline__ int aIdx(int k, int m) {
    return (((k >> 1) * MTILE + m) << 1) | (k & 1);
}

// ---------------------------------------------------------------------------
// Prep: w_def [64][576] -> wT2 pair-interleaved [288][64][2];
//       w_off [18][576] -> wToff2 zero-padded [288][32][2].
// ---------------------------------------------------------------------------
__global__ void prep_weights_kernel(const float* __restrict__ w_def,
                                    const float* __restrict__ w_off,
                                    float* __restrict__ wT2,
                                    float* __restrict__ wToff2) {
    int i = blockIdx.x * blockDim.x + threadIdx.x;
    if (i < KDIM * COUT) {
        int k = i / COUT, n = i % COUT;             // k = c*9 + tap
        wT2[(((k >> 1) * COUT + n) << 1) | (k & 1)] = w_def[n * KDIM + k];
    } else if (i < KDIM * COUT + KDIM * NOFFP) {
        int j = i - KDIM * COUT;
        int k = j / NOFFP, n = j % NOFFP;
        wToff2[(((k >> 1) * NOFFP + n) << 1) | (k & 1)] =
            (n < NOFF) ? w_off[n * KDIM + k] : 0.0f;
    }
}

// ---------------------------------------------------------------------------
__global__ void __launch_bounds__(128)
deform_fused_kernel(const float* __restrict__ x,
                    const float* __restrict__ wT2,     // [288][64][2]
                    const float* __restrict__ wToff2,  // [288][32][2]
                    const float* __restrict__ b_off,   // [18]
                    const float* __restrict__ b_def,   // [64]
                    float* __restrict__ out) {
    __shared__ float  ldsA[KDIM * MTILE];   // 36 KB, pair-interleaved
    __shared__ float  offsL[NOFF * MTILE];  // 18 x 16 offset values
    __shared__ int4   sIdx[NPAIR];          // clamped bilinear corner indices
    __shared__ float4 sWgt[NPAIR];          // bilinear weights (0 if OOB)

    const int bid = blockIdx.x;
    const int b   = bid / (HH * (WW / MTILE));
    const int rem = bid % (HH * (WW / MTILE));
    const int h   = rem / (WW / MTILE);
    const int w0  = (rem % (WW / MTILE)) * MTILE;
    const int tid  = threadIdx.x;
    const int wave = tid >> 5;
    const int lane = tid & 31;
    const int mA   = lane & 15;             // A/B frag: M/N = lane&15
    const int selp = lane >> 4;             // K pair sub-select: 0 or 1
    const int half = lane >> 4;

    const float* xb = x + (size_t)b * CIN * HW;
    const v2f* ldsA2   = (const v2f*)ldsA;
    const v2f* wT2v    = (const v2f*)wT2;
    const v2f* wToff2v = (const v2f*)wToff2;

    // ---- Phase I: regular im2col panel (zero-padded 3x3 taps) --------------
    for (int i = tid; i < CIN * NPAIR; i += 128) {   // 72 iterations
        const int c   = i / NPAIR;
        const int p   = i % NPAIR;
        const int tap = p >> 4;
        const int m   = p & 15;
        const int yy  = h + tap / 3 - 1;
        const int xx  = w0 + m + tap % 3 - 1;
        float v = 0.0f;
        if (yy >= 0 && yy < HH && xx >= 0 && xx < WW)
            v = xb[c * HW + yy * WW + xx];
        ldsA[aIdx(c * K2 + tap, m)] = v;
    }
    __syncthreads();

    // ---- Phase II: offset conv on the matrix pipe (waves 0-1) --------------
    if (wave < 2) {
        const int n0 = wave * 16;
        v8f oacc = {};
        for (int kb = 0; kb < KDIM; kb += 4) {
            const int kp = (kb >> 1) + selp;         // pair row
            v2f a  = ldsA2[kp * MTILE + mA];         // one ds_load_b64
            v2f bf = wToff2v[(size_t)kp * NOFFP + n0 + mA]; // one b64 load
            oacc = __builtin_amdgcn_wmma_f32_16x16x4_f32(
                       false, a, false, bf, (short)0, oacc, false, false);
        }
        const int o = n0 + (lane & 15);       // offset channel of this lane
        if (o < NOFF) {
            const float bias = b_off[o];
#pragma unroll
            for (int r = 0; r < 8; ++r)       // D layout: M = r + 8*half
                offsL[o * MTILE + (r + half * 8)] = oacc[r] + bias;
        }
    }
    __syncthreads();

    // ---- Phase III: bilinear corner data for 9 taps x 16 pixels ------------
    if (tid < NPAIR) {
        const int tap = tid >> 4;
        const int m   = tid & 15;
        const float dy = offsL[(2 * tap)     * MTILE + m];
        const float dx = offsL[(2 * tap + 1) * MTILE + m];
        const float ys = (float)(h + tap / 3 - 1) + dy;
        const float xs = (float)(w0 + m + tap % 3 - 1) + dx;
        const float y0f = floorf(ys), x0f = floorf(xs);
        const float wy = ys - y0f, wx = xs - x0f;
        const int y0 = (int)y0f, x0 = (int)x0f;
        const int y1 = y0 + 1,  x1 = x0 + 1;
        const bool vy0 = (y0 >= 0) && (y0 < HH);
        const bool vy1 = (y1 >= 0) && (y1 < HH);
        const bool vx0 = (x0 >= 0) && (x0 < WW);
        const bool vx1 = (x1 >= 0) && (x1 < WW);
        const int cy0 = min(max(y0, 0), HH - 1), cy1 = min(max(y1, 0), HH - 1);
        const int cx0 = min(max(x0, 0), WW - 1), cx1 = min(max(x1, 0), WW - 1);
        sIdx[tid] = make_int4(cy0 * WW + cx0, cy0 * WW + cx1,
                              cy1 * WW + cx0, cy1 * WW + cx1);
        sWgt[tid] = make_float4((vy0 && vx0) ? (1.0f - wy) * (1.0f - wx) : 0.0f,
                                (vy0 && vx1) ? (1.0f - wy) * wx          : 0.0f,
                                (vy1 && vx0) ? wy * (1.0f - wx)          : 0.0f,
                                (vy1 && vx1) ? wy * wx                   : 0.0f);
    }
    __syncthreads();

    // ---- Phase IV: gather sampled panel (overwrite ldsA) -------------------
    // i = p*64 + c, stride 128 => each thread keeps channel c = tid&63 fixed:
    // all 4 data-dependent gathers stay inside one 36 KB channel plane (L0-hot)
    for (int i = tid; i < NPAIR * CIN; i += 128) {
        const int p   = i >> 6;               // (tap, m) pair
        const int c   = i & 63;
        const int tap = p >> 4;
        const int m   = p & 15;
        const int4   ix = sIdx[p];
        const float4 wg = sWgt[p];
        const float* xc = xb + c * HW;
        const float v = wg.x * xc[ix.x] + wg.y * xc[ix.y]
                      + wg.z * xc[ix.z] + wg.w * xc[ix.w];
        ldsA[aIdx(c * K2 + tap, m)] = v;
    }
    __syncthreads();

    // ---- Phase V: main GEMM, 4 waves x 144 fp32 WMMAs ----------------------
    const int n0 = wave * 16;
    v8f acc = {};
    for (int kb = 0; kb < KDIM; kb += 4) {
        const int kp = (kb >> 1) + selp;
        v2f a  = ldsA2[kp * MTILE + mA];               // ds_load_b64
        v2f bf = wT2v[(size_t)kp * COUT + n0 + mA];    // global_load_b64
        acc = __builtin_amdgcn_wmma_f32_16x16x4_f32(
                  false, a, false, bf, (short)0, acc, false, false);
    }

    // ---- Epilogue: bias + two b128 stores (lane owns 8 consecutive w) ------
    const int n = n0 + (lane & 15);
    const float bias = b_def[n];
    float* op = out + (((size_t)b * COUT + n) * HH + h) * WW + w0 + half * 8;
    float4 lo = make_float4(acc[0] + bias, acc[1] + bias,
                            acc[2] + bias, acc[3] + bias);
    float4 hi = make_float4(acc[4] + bias, acc[5] + bias,
                            acc[6] + bias, acc[7] + bias);
    *(float4*)(op)     = lo;
    *(float4*)(op + 4) = hi;
}

// ---------------------------------------------------------------------------
extern "C" void kernel_launch(void* const* d_in, const int* in_sizes, int n_in,
                              void* d_out, int out_size, void* d_ws, size_t ws_size,
                              hipStream_t stream) {
    const float* x     = (const float*)d_in[0];
    const float* w_off = (const float*)d_in[1];
    const float* b_off = (const float*)d_in[2];
    const float* w_def = (const float*)d_in[3];
    const float* b_def = (const float*)d_in[4];
    float* out = (float*)d_out;

    // workspace: wT2 [288][64][2] then wToff2 [288][32][2]  (~221 KB total)
    float* wT2    = (float*)d_ws;
    float* wToff2 = wT2 + (size_t)KDIM * COUT;

    const int prepN = KDIM * COUT + KDIM * NOFFP;   // 55296
    prep_weights_kernel<<<(prepN + 255) / 256, 256, 0, stream>>>(w_def, w_off, wT2, wToff2);
    deform_fused_kernel<<<BB * HH * (WW / MTILE), 128, 0, stream>>>(
        x, wT2, wToff2, b_off, b_def, out);
}